// QLSTMGen_36498632081411
// MI455X (gfx1250) — compile-verified
//
#include <hip/hip_runtime.h>
#include <hip/hip_bf16.h>
#include <math.h>

// Problem constants (match reference)
#define B_  128
#define T_  512
#define I_  512
#define H_  512
#define NQ_ 8
#define K_  1024           // I_ + H_
#define NGATE 1536         // 3 * H_

typedef float v2f __attribute__((ext_vector_type(2)));
typedef float v8f __attribute__((ext_vector_type(8)));

__device__ __forceinline__ float fast_sigmoid(float x) {
    return 1.0f / (1.0f + __expf(-x));
}
__device__ __forceinline__ float fast_tanh(float x) {
    float e = __expf(-2.0f * x);
    return (1.0f - e) / (1.0f + e);
}

// ---------------------------------------------------------------------------
// Init: copy h0 -> h_ws, c0 -> c_ws
// ---------------------------------------------------------------------------
__global__ void qlstm_init_kernel(const float* __restrict__ h0,
                                  const float* __restrict__ c0,
                                  float* __restrict__ h_ws,
                                  float* __restrict__ c_ws) {
    int idx = blockIdx.x * blockDim.x + threadIdx.x;
    if (idx < B_ * H_) {
        h_ws[idx] = h0[idx];
        c_ws[idx] = c0[idx];
    }
}

// ---------------------------------------------------------------------------
// Per-timestep GEMM: P[gate][b][n] = act( z @ W_gate + b_gate )
//   z = concat(x[:,t,:], h_prev)  -> [B_, K_]
//   gates: 0=i (sigmoid), 1=g (tanh), 2=o (sigmoid)
// Block: 64 threads (2 waves). Each block: 16 batch rows x 128 output cols.
// Each wave: 16x64 strip via 4x V_WMMA_F32_16X16X4_F32 accumulators.
// Grid: (12, 8) = (NGATE/128, B_/16)  -> 96 workgroups/step for WGP coverage.
// ---------------------------------------------------------------------------
#define ZS_STRIDE 1028   // 1024 + 4 pad: row r, col k -> bank (4r + k) % 64, conflict-free

__global__ void __launch_bounds__(64)
qlstm_gemm_kernel(const float* __restrict__ x,
                  const float* __restrict__ h_prev,
                  const float* __restrict__ Wi, const float* __restrict__ bi,
                  const float* __restrict__ Wg, const float* __restrict__ bg,
                  const float* __restrict__ Wo, const float* __restrict__ bo,
                  float* __restrict__ P,
                  int t) {
    __shared__ float zs[16 * ZS_STRIDE];

    const int tid  = threadIdx.x;
    const int wave = tid >> 5;          // 0..1
    const int lane = tid & 31;
    const int lane_lo = lane & 15;      // M row (A) / N col (B) within tile
    const int hi      = lane >> 4;      // 0: K=0,1   1: K=2,3

    const int m0 = blockIdx.y * 16;     // batch-row tile base

    // ---- Stage z tile (16 rows x 1024 cols) into LDS --------------------
    // 64 threads: thread -> (row = tid/4, 256-col chunk = (tid%4)*256).
    // Chunks never straddle the x/h split (256 | 512).
    {
        const int row    = tid >> 2;
        const int cbase  = (tid & 3) * 256;
        float* dst = &zs[row * ZS_STRIDE + cbase];
        const float* src;
        if (cbase < I_) {
            src = x + (size_t)(m0 + row) * T_ * I_ + (size_t)t * I_ + cbase;
        } else {
            src = h_prev + (size_t)(m0 + row) * H_ + (cbase - I_);
        }
        #pragma unroll
        for (int i = 0; i < 256; i += 4) {
            float4 v = *(const float4*)(src + i);
            *(float4*)(dst + i) = v;
        }
    }
    __syncthreads();

    // ---- Select gate + weight/bias for this wave's 64-col strip ---------
    const int n_global  = blockIdx.x * 128 + wave * 64;   // 0..1535
    const int gate      = n_global >> 9;                  // 0,1,2 (64 | 512)
    const int n_in_gate = n_global & 511;

    const float* Wsel = (gate == 0) ? Wi : (gate == 1) ? Wg : Wo;
    const float* bsel = (gate == 0) ? bi : (gate == 1) ? bg : bo;

    // Per-lane B-column pointers, pre-offset by hi's K pair (rows k0+2hi, k0+2hi+1)
    const float* wcol0 = Wsel + (n_in_gate +  0 + lane_lo) + (size_t)(2 * hi) * H_;
    const float* wcol1 = Wsel + (n_in_gate + 16 + lane_lo) + (size_t)(2 * hi) * H_;
    const float* wcol2 = Wsel + (n_in_gate + 32 + lane_lo) + (size_t)(2 * hi) * H_;
    const float* wcol3 = Wsel + (n_in_gate + 48 + lane_lo) + (size_t)(2 * hi) * H_;

    // A fragment source: row = lane_lo, K pair selected by hi
    const float* zbase = &zs[lane_lo * ZS_STRIDE + 2 * hi];

    v8f acc0 = {0.f,0.f,0.f,0.f,0.f,0.f,0.f,0.f};
    v8f acc1 = {0.f,0.f,0.f,0.f,0.f,0.f,0.f,0.f};
    v8f acc2 = {0.f,0.f,0.f,0.f,0.f,0.f,0.f,0.f};
    v8f acc3 = {0.f,0.f,0.f,0.f,0.f,0.f,0.f,0.f};

    #pragma unroll 4
    for (int k0 = 0; k0 < K_; k0 += 4) {
        // A: 16x4 fp32 fragment (2 VGPRs/lane) from LDS (ds_load_b64)
        v2f a = *(const v2f*)(zbase + k0);
        const size_t ro = (size_t)k0 * H_;

        v2f b0; b0.x = wcol0[ro]; b0.y = wcol0[ro + H_];
        v2f b1; b1.x = wcol1[ro]; b1.y = wcol1[ro + H_];
        v2f b2; b2.x = wcol2[ro]; b2.y = wcol2[ro + H_];
        v2f b3; b3.x = wcol3[ro]; b3.y = wcol3[ro + H_];

        acc0 = __builtin_amdgcn_wmma_f32_16x16x4_f32(false, a, false, b0, (short)0, acc0, false, false);
        acc1 = __builtin_amdgcn_wmma_f32_16x16x4_f32(false, a, false, b1, (short)0, acc1, false, false);
        acc2 = __builtin_amdgcn_wmma_f32_16x16x4_f32(false, a, false, b2, (short)0, acc2, false, false);
        acc3 = __builtin_amdgcn_wmma_f32_16x16x4_f32(false, a, false, b3, (short)0, acc3, false, false);
    }

    // ---- Fused bias + activation + store --------------------------------
    // D layout: VGPR r holds (M = r + 8*hi, N = n0 + lane_lo)
    float* Pg = P + (size_t)gate * (B_ * H_);
    #pragma unroll
    for (int s = 0; s < 4; ++s) {
        const v8f acc = (s == 0) ? acc0 : (s == 1) ? acc1 : (s == 2) ? acc2 : acc3;
        const int ncol = n_in_gate + s * 16 + lane_lo;
        const float bias = bsel[ncol];
        #pragma unroll
        for (int r = 0; r < 8; ++r) {
            const int m = m0 + r + 8 * hi;
            float v = acc[r] + bias;
            v = (gate == 1) ? fast_tanh(v) : fast_sigmoid(v);
            Pg[(size_t)m * H_ + ncol] = v;
        }
    }
}

// ---------------------------------------------------------------------------
// Per-timestep pointwise: quantum forget gate + recurrence.
// Block b (128 blocks) owns batch row b; 512 threads.
//   q[j] = sum_k z[b,k] * Wf[k,j] + bf[j]        (LDS tree reduction)
//   q <- tanh(q*a_k + b_k) x3 ; q += qparams
//   f[n] = sigmoid( sum_j q[j]*Wfp[j,n] + bfp[n] )
//   c = f*c + i*g ; h = o*tanh(c) ; out[b,t,:] = h
// ---------------------------------------------------------------------------
__global__ void __launch_bounds__(512)
qlstm_point_kernel(const float* __restrict__ x,
                   float* __restrict__ h_ws,
                   float* __restrict__ c_ws,
                   const float* __restrict__ P,
                   const float* __restrict__ Wf, const float* __restrict__ bf,
                   const float* __restrict__ qnn_a, const float* __restrict__ qnn_b,
                   const float* __restrict__ qparams,
                   const float* __restrict__ Wfp, const float* __restrict__ bfp,
                   float* __restrict__ out,
                   int t) {
    __shared__ float red[512 * NQ_];
    __shared__ float tq[NQ_];

    const int b   = blockIdx.x;
    const int tid = threadIdx.x;

    // z contributions: k1 = tid (from x), k2 = tid + 512 (from h_prev)
    const float z1 = x[(size_t)b * T_ * I_ + (size_t)t * I_ + tid];
    const float z2 = h_ws[(size_t)b * H_ + tid];

    #pragma unroll
    for (int j = 0; j < NQ_; ++j) {
        red[tid * NQ_ + j] = z1 * Wf[(size_t)tid * NQ_ + j]
                           + z2 * Wf[(size_t)(tid + I_) * NQ_ + j];
    }
    __syncthreads();

    for (int s = 256; s > 0; s >>= 1) {
        if (tid < s) {
            #pragma unroll
            for (int j = 0; j < NQ_; ++j)
                red[tid * NQ_ + j] += red[(tid + s) * NQ_ + j];
        }
        __syncthreads();
    }

    if (tid < NQ_) {
        float q = red[tid] + bf[tid];
        #pragma unroll
        for (int k = 0; k < 3; ++k)
            q = fast_tanh(q * qnn_a[k] + qnn_b[k]);
        tq[tid] = q + qparams[tid];
    }
    __syncthreads();

    // Back-projection + recurrence for output column n = tid
    const int n = tid;
    float s = bfp[n];
    #pragma unroll
    for (int j = 0; j < NQ_; ++j)
        s += tq[j] * Wfp[(size_t)j * H_ + n];
    const float f = fast_sigmoid(s);

    const size_t bh = (size_t)b * H_ + n;
    const float i_g = P[0 * (size_t)(B_ * H_) + bh];   // already sigmoid()
    const float g_g = P[1 * (size_t)(B_ * H_) + bh];   // already tanh()
    const float o_g = P[2 * (size_t)(B_ * H_) + bh];   // already sigmoid()

    const float c_new = f * c_ws[bh] + i_g * g_g;
    const float h_new = o_g * fast_tanh(c_new);

    c_ws[bh] = c_new;
    h_ws[bh] = h_new;
    out[(size_t)b * T_ * H_ + (size_t)t * H_ + n] = h_new;
}

// ---------------------------------------------------------------------------
// Launch: init, then T_ x (GEMM kernel ; pointwise kernel), stream-ordered.
// ---------------------------------------------------------------------------
extern "C" void kernel_launch(void* const* d_in, const int* in_sizes, int n_in,
                              void* d_out, int out_size, void* d_ws, size_t ws_size,
                              hipStream_t stream) {
    const float* x      = (const float*)d_in[0];
    const float* h0     = (const float*)d_in[1];
    const float* c0     = (const float*)d_in[2];
    const float* Wi     = (const float*)d_in[3];
    const float* bi     = (const float*)d_in[4];
    const float* Wg     = (const float*)d_in[5];
    const float* bg     = (const float*)d_in[6];
    const float* Wo     = (const float*)d_in[7];
    const float* bo     = (const float*)d_in[8];
    const float* Wf     = (const float*)d_in[9];
    const float* bf     = (const float*)d_in[10];
    const float* qnn_a  = (const float*)d_in[11];
    const float* qnn_b  = (const float*)d_in[12];
    const float* qparams= (const float*)d_in[13];
    const float* Wfp    = (const float*)d_in[14];
    const float* bfp    = (const float*)d_in[15];
    float* out = (float*)d_out;

    // Workspace layout (floats): h_ws | c_ws | P[3][B][H]
    float* h_ws = (float*)d_ws;
    float* c_ws = h_ws + (size_t)B_ * H_;
    float* P    = c_ws + (size_t)B_ * H_;

    qlstm_init_kernel<<<(B_ * H_ + 255) / 256, 256, 0, stream>>>(h0, c0, h_ws, c_ws);

    for (int t = 0; t < T_; ++t) {
        qlstm_gemm_kernel<<<dim3(NGATE / 128, B_ / 16), 64, 0, stream>>>(
            x, h_ws, Wi, bi, Wg, bg, Wo, bo, P, t);
        qlstm_point_kernel<<<dim3(B_), 512, 0, stream>>>(
            x, h_ws, c_ws, P, Wf, bf, qnn_a, qnn_b, qparams, Wfp, bfp, out, t);
    }
}